// NAFLayer_35871566856789
// MI455X (gfx1250) — compile-verified
//
#include <hip/hip_runtime.h>
#include <math.h>

typedef __attribute__((ext_vector_type(2))) float v2f;
typedef __attribute__((ext_vector_type(8))) float v8f;

#define NB_ACTIONS 64
#define NB_ELEMS   2080            // 64*65/2
#define WAVES      4               // waves (batches) per block
#define SLICE      (NB_ELEMS + NB_ACTIONS)  // 2080 L + 64 d = 2144 floats

// One wave32 processes one batch row:
//   quad = || L^T d ||^2 , out = -0.5*quad
__global__ __launch_bounds__(WAVES * 32)
void naf_quad_kernel(const float* __restrict__ Lflat,
                     const float* __restrict__ mu,
                     const float* __restrict__ a,
                     float* __restrict__ out, int Btotal)
{
    __shared__ float lds[WAVES * SLICE];

    const int lane = threadIdx.x & 31;
    const int w    = threadIdx.x >> 5;
    int b = blockIdx.x * WAVES + w;
    if (b >= Btotal) b = Btotal - 1;   // duplicate work, identical value: safe

    float* Ls = &lds[w * SLICE];       // packed lower-tri row
    float* Ld = Ls + NB_ELEMS;         // d vector (64)

    // ---- phase 1: stream L row (8320 B) into LDS, coalesced b128 ----
    const float4* src = (const float4*)(Lflat + (size_t)b * NB_ELEMS);
    float4* dst = (float4*)Ls;
    #pragma unroll
    for (int t = 0; t < 16; ++t)
        dst[t * 32 + lane] = src[t * 32 + lane];          // 2048 floats
    Ls[2048 + lane] = Lflat[(size_t)b * NB_ELEMS + 2048 + lane]; // tail 32

    {   // d = a - mu (float2 per lane)
        const float2 m2 = ((const float2*)(mu + (size_t)b * NB_ACTIONS))[lane];
        const float2 a2 = ((const float2*)(a  + (size_t)b * NB_ACTIONS))[lane];
        float2 dd; dd.x = a2.x - m2.x; dd.y = a2.y - m2.y;
        ((float2*)Ld)[lane] = dd;
    }
    __syncthreads();

    // ---- phase 2: diagonal entries -> exp(x) + eps (2 per lane) ----
    #pragma unroll
    for (int t = 0; t < 2; ++t) {
        int j   = lane + t * 32;
        int idx = (j * (j + 3)) >> 1;      // i*(i+1)/2 + i
        Ls[idx] = expf(Ls[idx]) + 1e-7f;
    }
    __syncthreads();

    // ---- phase 3: y = L^T d via V_WMMA_F32_16X16X4_F32 (triangular tiles) ----
    // A (16x4) tile of L^T: lanes 0-15 hold M=lane&15, K=0(v0)/1(v1);
    //                       lanes 16-31 hold K=2/3.
    // B (4x16): d chunk broadcast across N -> per-half-wave constant.
    // Tile (kb, mb): zero when kb < 4*mb (skipped, 24), diagonal-crossing when
    // mb == kb>>2 (masked, 16), else strictly interior (plain loads, 24).
    const int half  = lane >> 4;       // 0: K=0,1   1: K=2,3
    const int mlane = lane & 15;
    const int h8    = half << 3;       // 8*half: induction increment term

    // Diagonal-tile masks depend only on r = kb&3 (8 compares total):
    //   i0 >= kcol  <=>  4r + 2*half     >= mlane
    //   i1 >= kcol  <=>  4r + 2*half + 1 >= mlane
    bool c0[4], c1[4];
    #pragma unroll
    for (int r = 0; r < 4; ++r) {
        c0[r] = (4 * r + 2 * half)     >= mlane;
        c1[r] = (4 * r + 2 * half + 1) >= mlane;
    }

    v8f acc[4];
    #pragma unroll
    for (int mb = 0; mb < 4; ++mb) acc[mb] = (v8f){};

    // Induction bases: b0 = i0*(i0+1)/2, b1 = i1*(i1+1)/2 with i0=4kb+2h, i1=i0+1
    int b0 = half ? 3 : 0;
    int b1 = half ? 6 : 1;

    #pragma unroll
    for (int kb = 0; kb < 16; ++kb) {
        const int i0e = kb * 4 + 2 * half;   // d index for B fragment
        v2f bfrag;
        bfrag.x = Ld[i0e];
        bfrag.y = Ld[i0e + 1];

        const int mbmax = kb >> 2;
        const int r     = kb & 3;
        #pragma unroll
        for (int mb = 0; mb <= mbmax; ++mb) {
            const int off = mb * 16 + mlane;     // kcol
            // masked addresses stay in [0,2080): max = 2016+63 = 2079
            float r0 = Ls[b0 + off];
            float r1 = Ls[b1 + off];
            v2f afrag;
            if (mb < mbmax) {          // strictly interior: i >= kcol always
                afrag.x = r0;
                afrag.y = r1;
            } else {                   // diagonal-crossing: hoisted masks
                afrag.x = c0[r] ? r0 : 0.0f;
                afrag.y = c1[r] ? r1 : 0.0f;
            }
            acc[mb] = __builtin_amdgcn_wmma_f32_16x16x4_f32(
                false, afrag, false, bfrag, (short)0, acc[mb], false, false);
        }

        // advance triangular bases: delta0 = 4*i0+10, delta1 = 4*i1+10
        b0 += 16 * kb + 10 + h8;
        b1 += 16 * kb + 14 + h8;
    }

    // ---- phase 4: quad = sum_k y_k^2 ; each y replicated across 16 lanes ----
    float s = 0.0f;
    #pragma unroll
    for (int mb = 0; mb < 4; ++mb)
        #pragma unroll
        for (int v = 0; v < 8; ++v)
            s += acc[mb][v] * acc[mb][v];

    #pragma unroll
    for (int off = 16; off > 0; off >>= 1)
        s += __shfl_xor(s, off, 32);

    if (lane == 0)
        out[b] = -0.03125f * s;   // -0.5 * (s / 16)
}

extern "C" void kernel_launch(void* const* d_in, const int* in_sizes, int n_in,
                              void* d_out, int out_size, void* d_ws, size_t ws_size,
                              hipStream_t stream) {
    const float* Lflat = (const float*)d_in[0];
    const float* mu    = (const float*)d_in[1];
    const float* a     = (const float*)d_in[2];
    float* out = (float*)d_out;

    const int B = in_sizes[1] / NB_ACTIONS;   // mu has B*64 elements
    const int blocks = (B + WAVES - 1) / WAVES;
    naf_quad_kernel<<<blocks, WAVES * 32, 0, stream>>>(Lflat, mu, a, out, B);
}